// HyperLayer_49649821942364
// MI455X (gfx1250) — compile-verified
//
#include <hip/hip_runtime.h>
#include <hip/hip_bf16.h>

#define BATCH   32
#define NPTS    262144
#define DIM_IN  8192
#define DIM_OUT 8192
#define CHUNKS  16
#define TPB     256

typedef unsigned int u32;
typedef u32   __attribute__((ext_vector_type(4))) u32x4;
typedef int   __attribute__((ext_vector_type(4))) i32x4;
typedef int   __attribute__((ext_vector_type(8))) i32x8;
typedef float __attribute__((ext_vector_type(4))) f32x4;
typedef float __attribute__((ext_vector_type(2))) f32x2;

// ---------------------------------------------------------------------------
// Kernel 1: out = bias  (d_out is poisoned by the harness, so seed it first)
// ---------------------------------------------------------------------------
__global__ void __launch_bounds__(TPB) hyper_init_bias(const float* __restrict__ bias,
                                                       float* __restrict__ out) {
    int i = blockIdx.x * TPB + threadIdx.x;            // one float4 per thread
    ((f32x4*)out)[i] = ((const f32x4*)bias)[i];
}

// ---------------------------------------------------------------------------
// Kernel 2: LDS-privatized gather/scatter.
//   - x[b] (32KB) staged into LDS via the Tensor Data Mover (gfx1250 TDM)
//   - partial y[b] (32KB) accumulated with ds_add_f32 in LDS
//   - flushed with global_atomic_add_f32
// ---------------------------------------------------------------------------
__global__ void __launch_bounds__(TPB) hyper_scatter(const float* __restrict__ x,
                                                     const float* __restrict__ ridx,
                                                     const float* __restrict__ rval,
                                                     float* __restrict__ out) {
    __shared__ float lds_x[DIM_IN];    // 32 KB: dense input vector for this batch
    __shared__ float lds_y[DIM_OUT];   // 32 KB: private output accumulator

    const int b     = blockIdx.y;
    const int chunk = blockIdx.x;
    const int tid   = threadIdx.x;

    // Zero the private accumulator with b128 stores (2048 float4 / 256 threads).
    {
        f32x4* y4 = (f32x4*)lds_y;
        const f32x4 z = {0.0f, 0.0f, 0.0f, 0.0f};
        #pragma unroll
        for (int j = tid; j < DIM_OUT / 4; j += TPB) y4[j] = z;
    }

    const float* xb = x + (size_t)b * DIM_IN;

#if defined(__gfx1250__) && __has_builtin(__builtin_amdgcn_tensor_load_to_lds)
    // --- Tensor Data Mover: DMA 8192 f32 (32KB) global -> LDS, one wave issues.
    if (tid == 0) {
        // Low 32 bits of the generic address of an LDS pointer == LDS byte offset.
        u32 lds_off = (u32)(uintptr_t)(void*)lds_x;
        unsigned long long ga = (unsigned long long)(uintptr_t)xb;

        u32x4 g0;
        g0.x = 1u;                                        // count=1 (valid), user mode
        g0.y = lds_off;                                   // lds_addr (bytes)
        g0.z = (u32)(ga & 0xFFFFFFFFu);                   // global_addr[31:0]
        g0.w = (u32)((ga >> 32) & 0x01FFFFFFu)            // global_addr[56:32]
             | (2u << 30);                                // type = 2 ("image")

        i32x8 g1;
        g1[0] = (int)(2u << 16);                          // data_size=2 -> 4 bytes; mask=0
        g1[1] = (int)((u32)(DIM_IN & 0xFFFF) << 16);      // tensor_dim0[15:0] @ bits[63:48]
        g1[2] = (int)(1u << 16);                          // tensor_dim1 = 1
        g1[3] = (int)((u32)(DIM_IN & 0xFFFF) << 16);      // tile_dim0 = 8192 @ bits[127:112]
        g1[4] = 1;                                        // tile_dim1 = 1, tile_dim2 = 0
        g1[5] = DIM_IN;                                   // tensor_dim0_stride[31:0]
        g1[6] = (int)((u32)(DIM_IN & 0xFFFF) << 16);      // tensor_dim1_stride[15:0]
        g1[7] = 0;

        i32x4 g2 = {0, 0, 0, 0};                          // dims 2..4 unused (tile_dim*=0)
        i32x4 g3 = {0, 0, 0, 0};
        i32x8 g4 = {0, 0, 0, 0, 0, 0, 0, 0};              // extra group (clang-23 6-arg form)

        __builtin_amdgcn_tensor_load_to_lds(g0, g1, g2, g3, g4, 0);
        __builtin_amdgcn_s_wait_tensorcnt(0);             // s_wait_tensorcnt 0
    }
#else
    for (int j = tid; j < DIM_IN; j += TPB) lds_x[j] = xb[j];
#endif
    __syncthreads();

    // --- Stream this chunk's entries: coalesced non-temporal b128/b64 loads.
    const int    per_chunk = NPTS / CHUNKS;                       // 16384 entries
    const size_t base      = (size_t)b * NPTS + (size_t)chunk * per_chunk;
    const f32x4* ridx4     = (const f32x4*)ridx;                  // 2 entries / float4
    const f32x2* rv2       = (const f32x2*)(rval + base);
    const size_t  pairBase = base >> 1;
    const int     pairs    = per_chunk / 2;                       // 8192 pairs

    for (int i = tid; i < pairs; i += TPB) {
        f32x4 r = __builtin_nontemporal_load(&ridx4[pairBase + i]);  // read-once stream
        f32x2 v = __builtin_nontemporal_load(&rv2[i]);

        int f0a = (int)floorf(r.x), c0a = (int)ceilf(r.x);
        int f1a = (int)floorf(r.y), c1a = (int)ceilf(r.y);
        int f0b = (int)floorf(r.z), c0b = (int)ceilf(r.z);
        int f1b = (int)floorf(r.w), c1b = (int)ceilf(r.w);

        // Each real index pair contributes s to BOTH out corners (4 corners total,
        // value repeated; degenerate floor==ceil double-counts, matching reference).
        float sa = v.x * (lds_x[f1a] + lds_x[c1a]);
        float sb = v.y * (lds_x[f1b] + lds_x[c1b]);

        unsafeAtomicAdd(&lds_y[f0a], sa);   // ds_add_f32 (no return)
        unsafeAtomicAdd(&lds_y[c0a], sa);
        unsafeAtomicAdd(&lds_y[f0b], sb);
        unsafeAtomicAdd(&lds_y[c0b], sb);
    }
    __syncthreads();

    // --- Flush private accumulator into global (L2-resident, 1 MiB total).
    float* outb = out + (size_t)b * DIM_OUT;
    #pragma unroll
    for (int j = tid; j < DIM_OUT; j += TPB) {
        unsafeAtomicAdd(&outb[j], lds_y[j]);   // global_atomic_add_f32
    }
}

// ---------------------------------------------------------------------------
extern "C" void kernel_launch(void* const* d_in, const int* in_sizes, int n_in,
                              void* d_out, int out_size, void* d_ws, size_t ws_size,
                              hipStream_t stream) {
    const float* x    = (const float*)d_in[0];   // (B, DIM_IN)
    const float* ridx = (const float*)d_in[1];   // (B, N, 2)
    const float* rval = (const float*)d_in[2];   // (B, N)
    const float* bias = (const float*)d_in[3];   // (B, DIM_OUT)
    float* out = (float*)d_out;                  // (B, DIM_OUT)

    // out = bias  (262144 floats -> 65536 float4 -> 256 blocks)
    hyper_init_bias<<<(BATCH * DIM_OUT) / (4 * TPB), TPB, 0, stream>>>(bias, out);

    // Scatter: 16 chunks x 32 batches = 512 workgroups, 64KB LDS each.
    dim3 grid(CHUNKS, BATCH, 1);
    hyper_scatter<<<grid, TPB, 0, stream>>>(x, ridx, rval, out);
}